// ZeroForwardLCASLoss_81638738362600
// MI455X (gfx1250) — compile-verified
//
#include <hip/hip_runtime.h>
#include <hip/hip_bf16.h>
#include <math.h>

typedef float v2f __attribute__((ext_vector_type(2)));
typedef float v4f __attribute__((ext_vector_type(4)));
typedef float v8f __attribute__((ext_vector_type(8)));

#define B_     8
#define M_     64
#define H_     1024
#define HEADS_ 16
#define K_     4096
#define J_     32

// gfx1250 async global->LDS path, compile-safe guarded
#if defined(__has_builtin)
#if __has_builtin(__builtin_amdgcn_global_load_async_to_lds_b32) && \
    __has_builtin(__builtin_amdgcn_s_wait_asynccnt)
#define USE_ASYNC_LDS 1
#endif
#endif

// workspace layout (float element offsets)
#define WS_HIDDEN 0
#define WS_IMP    (WS_HIDDEN + B_*M_*H_)       // 524288 + 512
#define WS_ANCH   (WS_IMP    + B_*M_)          // + 8192
#define WS_W      (WS_ANCH   + B_*H_)          // + 32768
#define WS_REPS   (WS_W      + B_*K_)          // + 262144
#define WS_RNORM  (WS_REPS   + B_*J_*H_)       // + 256
#define WS_ANORM  (WS_RNORM  + B_*J_)          // + 8
#define WS_S      (WS_ANORM  + B_)             // + 2048

// ---------------------------------------------------------------- K0: zero w
__global__ __launch_bounds__(256)
void k0_zero(float* __restrict__ p, int n) {
    int i = blockIdx.x * 256 + threadIdx.x;
    if (i < n) p[i] = 0.0f;
}

// ------------------------------------------- K1: hidden = tanh(A @ W1 + b1)
// A = prefix_slots viewed as [B*M, H] row-major, W1 [H,H] row-major.
// One wave computes FOUR 16x16 row tiles for one column tile, reusing each
// B fragment across 4 V_WMMA_F32_16X16X4_F32 (cuts W1 re-reads 4x).
__global__ __launch_bounds__(128)
void k1_hidden_wmma(const float* __restrict__ A, const float* __restrict__ W1,
                    const float* __restrict__ b1, float* __restrict__ hidden) {
    const int lane = threadIdx.x;                  // 0..31 (wave32)
    const int tm0  = blockIdx.x * 4;               // row tiles tm0..tm0+3 (of 32)
    const int tn   = blockIdx.y * 4 + threadIdx.y; // 0..63 column tiles
    const int half = lane >> 4;                    // 0 | 1
    const int lm   = lane & 15;
    const int col  = tn * 16 + lm;

    // A frag: lane lm holds row, K pair at k + 2*half   (ISA 7.12.2)
    const float* ap[4];
    #pragma unroll
    for (int t = 0; t < 4; ++t)
        ap[t] = A + (size_t)((tm0 + t) * 16 + lm) * H_ + 2 * half;
    // B frag: VGPR0 = row K (+2 for high half), VGPR1 = row K+1, N = lm
    const float* bptr = W1 + (size_t)(2 * half) * H_ + col;

    v8f c[4] = {};
    for (int kk = 0; kk < H_; kk += 4) {
        v2f b;
        b.x = bptr[(size_t)kk * H_];
        b.y = bptr[(size_t)(kk + 1) * H_];
        #pragma unroll
        for (int t = 0; t < 4; ++t) {
            v2f a = *(const v2f*)(ap[t] + kk);     // 8B aligned pair
            c[t] = __builtin_amdgcn_wmma_f32_16x16x4_f32(
                       /*neg_a=*/false, a, /*neg_b=*/false, b,
                       /*c_mod=*/(short)0, c[t], /*reuse_a=*/false, /*reuse_b=*/false);
        }
    }

    const float bias = b1[col];
    #pragma unroll
    for (int t = 0; t < 4; ++t) {
        #pragma unroll
        for (int v = 0; v < 8; ++v) {              // C/D: VGPR v -> row v (+8 hi)
            int row = (tm0 + t) * 16 + v + 8 * half;
            hidden[(size_t)row * H_ + col] = tanhf(c[t][v] + bias);
        }
    }
}

// ------------------- K2: gate = hidden@W2 + b2 ; imp = softmax(sigmoid(gate))
__global__ __launch_bounds__(64)
void k2_imp(const float* __restrict__ hidden, const float* __restrict__ W2,
            const float* __restrict__ b2, float* __restrict__ imp) {
    __shared__ float g[M_];
    const int b = blockIdx.x, m = threadIdx.x;
    const float* row = hidden + (size_t)(b * M_ + m) * H_;
    float acc = 0.0f;
    for (int h = 0; h < H_; ++h) acc += row[h] * W2[h];
    const float sig = 1.0f / (1.0f + __expf(-(acc + b2[0])));
    g[m] = sig;
    __syncthreads();
    float mx = -1e30f;
    for (int i = 0; i < M_; ++i) mx = fmaxf(mx, g[i]);
    float sum = 0.0f;
    for (int i = 0; i < M_; ++i) sum += __expf(g[i] - mx);
    imp[b * M_ + m] = __expf(sig - mx) / sum;
}

// ----------------------------------- K3: anchors[b,h] = sum_m imp * prefix
__global__ __launch_bounds__(256)
void k3_anchors(const float* __restrict__ prefix, const float* __restrict__ imp,
                float* __restrict__ anchors) {
    const int b = blockIdx.y;
    const int h = blockIdx.x * 256 + threadIdx.x;
    float acc = 0.0f;
    for (int m = 0; m < M_; ++m)
        acc += imp[b * M_ + m] * prefix[(size_t)(b * M_ + m) * H_ + h];
    anchors[b * H_ + h] = acc;
}

// ------- K4: w[b,k] = sum_{h,m} (imp[b,m]/heads) * cw[b,h,m,k]  (128 MB pass)
// float4 per thread -> global_load_b128; each block streams 4 KB/iteration.
__global__ __launch_bounds__(256)
void k4_w(const float* __restrict__ cw, const float* __restrict__ imp,
          float* __restrict__ w) {
    __shared__ float coef[M_];
    const int b = blockIdx.z, hmc = blockIdx.y, ktile = blockIdx.x; // ktile 0..3
    if (threadIdx.x < M_)
        coef[threadIdx.x] = imp[b * M_ + threadIdx.x] * (1.0f / HEADS_);
    __syncthreads();
    const int k4 = ktile * 256 + threadIdx.x;       // float4 column, 0..1023
    const v4f* base =
        (const v4f*)(cw + ((size_t)b * HEADS_ * M_ + (size_t)hmc * 128) * K_) + k4;
    v4f acc = {};
    #pragma unroll 4
    for (int i = 0; i < 128; ++i) {                 // hm = h*M_ + m row-major
        const float cf = coef[(hmc * 128 + i) & (M_ - 1)];
        const v4f x = base[(size_t)i * (K_ / 4)];
        acc.x += cf * x.x; acc.y += cf * x.y;
        acc.z += cf * x.z; acc.w += cf * x.w;
    }
    float* wp = w + b * K_ + k4 * 4;
    atomicAdd(wp + 0, acc.x);
    atomicAdd(wp + 1, acc.y);
    atomicAdd(wp + 2, acc.z);
    atomicAdd(wp + 3, acc.w);
}

// --------------- K5: reps[b,j,h] = sum_{k in span} w[b,k] * kv[b,k,h]
// One block per (b,j); thread owns 4 contiguous h (b128 loads/stores).
// Span weights staged into LDS (async global->LDS on gfx1250 when available).
__global__ __launch_bounds__(256)
void k5_reps(const float* __restrict__ kv, const float* __restrict__ w,
             const int* __restrict__ spans, float* __restrict__ reps) {
    __shared__ float wl[128];
    const int j = blockIdx.x, b = blockIdx.y;
    int s = spans[(b * J_ + j) * 2 + 0];
    int e = spans[(b * J_ + j) * 2 + 1];
    s = min(max(s, 0), K_);
    e = max(min(max(e, 0), K_), s);
    const int len = min(e - s, 128);                // <=128 by construction

    if (threadIdx.x < 128) {                        // waves 0-3 fully active
        int idx = s + (int)threadIdx.x;
        if (idx > K_ - 1) idx = K_ - 1;             // stay in-bounds; tail unused
#if USE_ASYNC_LDS
        __builtin_amdgcn_global_load_async_to_lds_b32(
            (__attribute__((address_space(1))) int*)(w + b * K_ + idx),
            (__attribute__((address_space(3))) int*)&wl[threadIdx.x],
            /*offset=*/0, /*cpol=*/0);
#else
        wl[threadIdx.x] = w[b * K_ + idx];
#endif
    }
#if USE_ASYNC_LDS
    __builtin_amdgcn_s_wait_asynccnt(0);
#endif
    __syncthreads();

    const int h4 = threadIdx.x;                     // float4 column of H
    const v4f* kvb = (const v4f*)(kv + (size_t)b * K_ * H_) + h4;
    v4f acc = {};
    for (int i = 0; i < len; ++i) {
        const float ww = wl[i];
        const v4f x = kvb[(size_t)(s + i) * (H_ / 4)];
        acc.x += ww * x.x; acc.y += ww * x.y;
        acc.z += ww * x.z; acc.w += ww * x.w;
    }
    ((v4f*)(reps + (size_t)(b * J_ + j) * H_))[h4] = acc;
}

// -------------------------- K6a: clamped L2 norms of reps rows + anchors
__global__ __launch_bounds__(256)
void k6a_norms(const float* __restrict__ reps, const float* __restrict__ anchors,
               float* __restrict__ rnorm, float* __restrict__ anorm) {
    __shared__ float red[256];
    const int i = blockIdx.x;                       // 0..B*J-1 reps, then B anchors
    const float* row = (i < B_ * J_) ? (reps + (size_t)i * H_)
                                     : (anchors + (size_t)(i - B_ * J_) * H_);
    float acc = 0.0f;
    for (int h = threadIdx.x; h < H_; h += 256) { float x = row[h]; acc += x * x; }
    red[threadIdx.x] = acc; __syncthreads();
    for (int st = 128; st > 0; st >>= 1) {
        if (threadIdx.x < st) red[threadIdx.x] += red[threadIdx.x + st];
        __syncthreads();
    }
    if (threadIdx.x == 0) {
        const float n = fmaxf(sqrtf(red[0]), 1e-8f);
        if (i < B_ * J_) rnorm[i] = n; else anorm[i - B_ * J_] = n;
    }
}

// ------------- K6b: S[b,c,j] = cos(anchors[b], reps[c,j]) / temperature
__global__ __launch_bounds__(256)
void k6b_scores(const float* __restrict__ anchors, const float* __restrict__ reps,
                const float* __restrict__ anorm, const float* __restrict__ rnorm,
                const float* __restrict__ temp, float* __restrict__ S) {
    __shared__ float red[256];
    const int j = blockIdx.x, c = blockIdx.y, b = blockIdx.z;
    const float* an = anchors + b * H_;
    const float* rn = reps + (size_t)(c * J_ + j) * H_;
    float acc = 0.0f;
    for (int h = threadIdx.x; h < H_; h += 256) acc += an[h] * rn[h];
    red[threadIdx.x] = acc; __syncthreads();
    for (int st = 128; st > 0; st >>= 1) {
        if (threadIdx.x < st) red[threadIdx.x] += red[threadIdx.x + st];
        __syncthreads();
    }
    if (threadIdx.x == 0)
        S[(b * B_ + c) * J_ + j] = red[0] / (anorm[b] * rnorm[c * J_ + j] * temp[0]);
}

// ------------------------------------------ K6c: scalar InfoNCE-style loss
__global__ __launch_bounds__(64)
void k6c_loss(const float* __restrict__ S, const int* __restrict__ spans,
              const int* __restrict__ labels, float* __restrict__ out) {
    __shared__ float lossv[B_];
    __shared__ int   haspos[B_];
    const int b = threadIdx.x;
    if (b < B_) {
        float pos = 0.0f, negown = 0.0f, other = 0.0f;
        int npos = 0;
        for (int j = 0; j < J_; ++j) {
            int s = spans[(b * J_ + j) * 2], e = spans[(b * J_ + j) * 2 + 1];
            s = min(max(s, 0), K_); e = max(min(max(e, 0), K_), s);
            if (e <= s) continue;
            const float Ev = __expf(S[(b * B_ + b) * J_ + j]);
            if (labels[b * J_ + j] == 1) { pos += Ev; npos++; }
            else                         { negown += Ev; }
        }
        for (int c = 0; c < B_; ++c) {
            if (c == b) continue;
            for (int j = 0; j < J_; ++j) {
                int s = spans[(c * J_ + j) * 2], e = spans[(c * J_ + j) * 2 + 1];
                s = min(max(s, 0), K_); e = max(min(max(e, 0), K_), s);
                if (e > s) other += __expf(S[(b * B_ + c) * J_ + j]);
            }
        }
        haspos[b] = (npos > 0) ? 1 : 0;
        const float neg = negown + other;
        lossv[b] = (npos > 0) ? -logf(pos / (pos + neg + 1e-8f)) : 0.0f;
    }
    __syncthreads();
    if (threadIdx.x == 0) {
        float s = 0.0f; int n = 0;
        for (int bb = 0; bb < B_; ++bb) { if (haspos[bb]) { s += lossv[bb]; n++; } }
        out[0] = s / (float)max(n, 1);
    }
}

extern "C" void kernel_launch(void* const* d_in, const int* in_sizes, int n_in,
                              void* d_out, int out_size, void* d_ws, size_t ws_size,
                              hipStream_t stream) {
    const float* prefix = (const float*)d_in[0];
    const float* cw     = (const float*)d_in[1];
    const float* kv     = (const float*)d_in[2];
    const float* W1     = (const float*)d_in[3];
    const float* b1     = (const float*)d_in[4];
    const float* W2     = (const float*)d_in[5];
    const float* b2     = (const float*)d_in[6];
    const float* temp   = (const float*)d_in[7];
    const int*   spans  = (const int*)d_in[8];
    const int*   labels = (const int*)d_in[9];

    float* ws      = (float*)d_ws;
    float* hidden  = ws + WS_HIDDEN;
    float* imp     = ws + WS_IMP;
    float* anchors = ws + WS_ANCH;
    float* w       = ws + WS_W;
    float* reps    = ws + WS_REPS;
    float* rnorm   = ws + WS_RNORM;
    float* anorm   = ws + WS_ANORM;
    float* S       = ws + WS_S;
    float* out     = (float*)d_out;

    k0_zero<<<(B_ * K_ + 255) / 256, 256, 0, stream>>>(w, B_ * K_);
    k1_hidden_wmma<<<dim3(8, 16), dim3(32, 4), 0, stream>>>(prefix, W1, b1, hidden);
    k2_imp<<<B_, 64, 0, stream>>>(hidden, W2, b2, imp);
    k3_anchors<<<dim3(H_ / 256, B_), 256, 0, stream>>>(prefix, imp, anchors);
    k4_w<<<dim3(4, 8, B_), 256, 0, stream>>>(cw, imp, w);
    k5_reps<<<dim3(J_, B_), 256, 0, stream>>>(kv, w, spans, reps);
    k6a_norms<<<B_ * J_ + B_, 256, 0, stream>>>(reps, anchors, rnorm, anorm);
    k6b_scores<<<dim3(J_, B_, B_), 256, 0, stream>>>(anchors, reps, anorm, rnorm, temp, S);
    k6c_loss<<<1, 64, 0, stream>>>(S, spans, labels, out);
}